// DeformAttention_18648747999835
// MI455X (gfx1250) — compile-verified
//
#include <hip/hip_runtime.h>
#include <math.h>

typedef float v2f __attribute__((ext_vector_type(2)));
typedef float v8f __attribute__((ext_vector_type(8)));

#define D_   16
#define H_   32
#define W_   32
#define LEN  16384          // D_*H_*W_
#define NB   4
#define DIM  256
#define HEADS 8
#define PTS  4
#define DH   32
#define NQ   (NB * LEN)     // 65536 flat queries
#define QT   32             // queries per block (two 16-row WMMA tiles)

__device__ __forceinline__ v8f wmma_f32(v2f a, v2f b, v8f c) {
    // D = A(16x4 f32) * B(4x16 f32) + C(16x16 f32)
    return __builtin_amdgcn_wmma_f32_16x16x4_f32(
        /*neg_a=*/false, a, /*neg_b=*/false, b,
        /*c_mod=*/(short)0, c, /*reuse_a=*/false, /*reuse_b=*/false);
}

// K=256 GEMM panel: one B fetch feeds two A tiles (two accumulators).
// r0/r1 point at ldsX[row][half*2]; bp points at W + cg + half*2*LDW.
// LDW is a compile-time constant so all offsets fold into vmem immediates.
template<int LDW>
__device__ __forceinline__ void gemm2(const float* __restrict__ r0,
                                      const float* __restrict__ r1,
                                      const float* __restrict__ bp,
                                      v8f& C0, v8f& C1)
{
    #pragma unroll 8
    for (int kk = 0; kk < DIM; kk += 4) {
        v2f B;  B.x  = bp[(size_t)kk * LDW]; B.y  = bp[(size_t)kk * LDW + LDW];
        v2f A0; A0.x = r0[kk];               A0.y = r0[kk + 1];
        v2f A1; A1.x = r1[kk];               A1.y = r1[kk + 1];
        C0 = wmma_f32(A0, B, C0);
        C1 = wmma_f32(A1, B, C1);
    }
}

// ---------------------------------------------------------------------------
// Kernel 1: fused input projections (value / sampling-offset->pixel coords /
// attention logits) for a tile of 32 queries per block.
// ---------------------------------------------------------------------------
__global__ __launch_bounds__(256) void proj_kernel(
    const float* __restrict__ x, const float* __restrict__ vr,
    const float* __restrict__ Wv, const float* __restrict__ bv,
    const float* __restrict__ Woff, const float* __restrict__ boff,
    const float* __restrict__ Wattn, const float* __restrict__ battn,
    float* __restrict__ valueBuf, float* __restrict__ locBuf,
    float* __restrict__ attnBuf)
{
    __shared__ float ldsX[QT][260];   // 32x256 x-tile, padded (260 % 64 != 0)

    const int q0  = blockIdx.x << 5;       // flat query base (32-aligned)
    const int nb  = q0 >> 14;              // batch (LEN = 2^14)
    const int ql0 = q0 & (LEN - 1);        // query-in-batch base

    // cooperative 32x256 f32 tile load (float4 x 8 per thread)
    const float* xblk = x + (size_t)q0 * DIM;
    for (int i = threadIdx.x; i < 2048; i += 256) {
        const int row = i >> 6;
        const int c4  = i & 63;
        const float4 v = ((const float4*)(xblk + row * DIM))[c4];
        float* dst = &ldsX[row][c4 << 2];
        dst[0] = v.x; dst[1] = v.y; dst[2] = v.z; dst[3] = v.w;
    }
    __syncthreads();

    const int wave = threadIdx.x >> 5;
    const int lane = threadIdx.x & 31;
    const int half = lane >> 4;   // A/B lane split: K pair 0/1 vs 2/3
    const int lrow = lane & 15;

    const float* r0 = &ldsX[lrow][half << 1];        // A tile 0 (queries 0..15)
    const float* r1 = &ldsX[16 + lrow][half << 1];   // A tile 1 (queries 16..31)

    const float vr0 = vr[nb * 3 + 0];
    const float vr1 = vr[nb * 3 + 1];
    const float vr2 = vr[nb * 3 + 2];

    // 24 column tiles of 16: [0,16) value, [16,22) offsets, [22,24) attn
    for (int t = 0; t < 3; ++t) {
        const int ct = wave + (t << 3);
        v8f C0 = {}, C1 = {};

        if (ct < 16) {                                   // ---- value ----
            const int cg = (ct << 4) + lrow;
            const float* bp = Wv + cg + (half << 1) * 256;
            __builtin_prefetch(bp, 0, 1);
            gemm2<256>(r0, r1, bp, C0, C1);

            const int h = cg >> 5, dh = cg & 31;
            const float bb = bv[cg];
            float* vb = valueBuf + (((size_t)nb * HEADS + h) * LEN) * DH + dh;
            #pragma unroll
            for (int rt = 0; rt < 2; ++rt) {
                const v8f& C = rt ? C1 : C0;
                #pragma unroll
                for (int r = 0; r < 8; ++r) {
                    const int ql = ql0 + (rt << 4) + r + (half << 3);
                    vb[(size_t)ql * DH] = C[r] + bb;     // (n, head, len, dh)
                }
            }
        } else if (ct < 22) {                            // ---- offsets ----
            const int cg = ((ct - 16) << 4) + lrow;      // 0..95
            const float* bp = Woff + cg + (half << 1) * 96;
            __builtin_prefetch(bp, 0, 1);
            gemm2<96>(r0, r1, bp, C0, C1);

            const int h = cg / 12, rem = cg % 12;
            const int p = rem / 3, axis = rem % 3;
            const float bb = boff[cg];
            #pragma unroll
            for (int rt = 0; rt < 2; ++rt) {
                const v8f& C = rt ? C1 : C0;
                #pragma unroll
                for (int r = 0; r < 8; ++r) {
                    const int ql = ql0 + (rt << 4) + r + (half << 3);
                    const int dz = ql >> 10, yy = (ql >> 5) & 31, xx = ql & 31;
                    const float val = C[r] + bb;
                    // pix = ref*size + off*(size/norm) - 0.5 ; size/norm == 1
                    float pix;
                    if (axis == 0)      pix = (dz + 0.5f) / vr0 + val - 0.5f; // d
                    else if (axis == 1) pix = (xx + 0.5f) / vr1 + val - 0.5f; // x
                    else                pix = (yy + 0.5f) / vr2 + val - 0.5f; // y
                    const size_t q = (size_t)nb * LEN + ql;
                    locBuf[(q * HEADS + h) * 12 + p * 3 + axis] = pix;
                }
            }
        } else {                                         // ---- attn logits ----
            const int cg = ((ct - 22) << 4) + lrow;      // 0..31 = h*4+p
            const float* bp = Wattn + cg + (half << 1) * 32;
            gemm2<32>(r0, r1, bp, C0, C1);

            const float bb = battn[cg];
            #pragma unroll
            for (int rt = 0; rt < 2; ++rt) {
                const v8f& C = rt ? C1 : C0;
                #pragma unroll
                for (int r = 0; r < 8; ++r) {
                    const int ql = ql0 + (rt << 4) + r + (half << 3);
                    const size_t q = (size_t)nb * LEN + ql;
                    attnBuf[q * 32 + cg] = C[r] + bb;    // softmax next
                }
            }
        }
    }
}

// ---------------------------------------------------------------------------
// Kernel 2: softmax over P=4 for each (query, head)
// ---------------------------------------------------------------------------
__global__ __launch_bounds__(256) void softmax_kernel(float* __restrict__ attnBuf)
{
    const int tid = blockIdx.x * 256 + threadIdx.x;
    if (tid >= NQ * HEADS) return;
    float* a = attnBuf + (size_t)tid * 4;
    const float a0 = a[0], a1 = a[1], a2 = a[2], a3 = a[3];
    const float m  = fmaxf(fmaxf(a0, a1), fmaxf(a2, a3));
    const float e0 = __expf(a0 - m), e1 = __expf(a1 - m);
    const float e2 = __expf(a2 - m), e3 = __expf(a3 - m);
    const float inv = 1.0f / (e0 + e1 + e2 + e3);
    a[0] = e0 * inv; a[1] = e1 * inv; a[2] = e2 * inv; a[3] = e3 * inv;
}

// ---------------------------------------------------------------------------
// Kernel 3: trilinear gather + attention-weighted sum into LDS, then fused
// output projection GEMM (never materializes the 64MB attn-output in HBM).
// 32 queries per block.
// ---------------------------------------------------------------------------
__global__ __launch_bounds__(256) void sample_out_kernel(
    const float* __restrict__ valueBuf, const float* __restrict__ locBuf,
    const float* __restrict__ attnBuf, const float* __restrict__ Wout,
    const float* __restrict__ bout, float* __restrict__ out)
{
    __shared__ float ldsA[QT][260];   // 32 queries x 256 (head*Dh), padded

    const int q0   = blockIdx.x << 5;
    const int nb   = q0 >> 14;
    const int wave = threadIdx.x >> 5;
    const int lane = threadIdx.x & 31;   // lane == dh channel (Dh == 32)

    // Phase 1: 256 (q,head) pairs; one wave per pair, lane = dh -> 128B gathers
    for (int i = 0; i < QT; ++i) {
        const int pair = wave + (i << 3);
        const int qql  = pair >> 3;          // local query 0..31
        const int h    = pair & 7;
        const size_t q = (size_t)q0 + qql;
        const float* vbase = valueBuf + ((size_t)nb * HEADS + h) * LEN * DH + lane;
        const float* lp = locBuf  + (q * HEADS + h) * 12;
        const float* ap = attnBuf + (q * HEADS + h) * 4;

        float acc = 0.0f;
        #pragma unroll
        for (int p = 0; p < PTS; ++p) {
            const float aw = ap[p];
            const float pd = lp[p * 3 + 0];
            const float px = lp[p * 3 + 1];
            const float py = lp[p * 3 + 2];
            const float fd0 = floorf(pd), fy0 = floorf(py), fx0 = floorf(px);
            const float fd = pd - fd0, fy = py - fy0, fx = px - fx0;
            const int d0 = (int)fd0, y0 = (int)fy0, x0 = (int)fx0;
            #pragma unroll
            for (int c = 0; c < 8; ++c) {
                const int dd = (c >> 2) & 1, dy = (c >> 1) & 1, dx = c & 1;
                const int id = d0 + dd, iy = y0 + dy, ix = x0 + dx;
                if (id >= 0 && id < D_ && iy >= 0 && iy < H_ && ix >= 0 && ix < W_) {
                    const float w = (dd ? fd : 1.0f - fd) *
                                    (dy ? fy : 1.0f - fy) *
                                    (dx ? fx : 1.0f - fx);
                    const int fidx = (id << 10) + (iy << 5) + ix;
                    acc += aw * w * vbase[(size_t)fidx * DH];
                }
            }
        }
        ldsA[qql][h * DH + lane] = acc;
    }
    __syncthreads();

    // Phase 2: (32 x 256) @ W_out(256 x 256) via dual-tile f32 WMMA, + bias
    const int half = lane >> 4;
    const int lrow = lane & 15;
    const float* r0 = &ldsA[lrow][half << 1];
    const float* r1 = &ldsA[16 + lrow][half << 1];

    for (int t = 0; t < 2; ++t) {
        const int cg = ((wave + (t << 3)) << 4) + lrow;
        const float* bp = Wout + cg + (half << 1) * 256;
        __builtin_prefetch(bp, 0, 1);
        v8f C0 = {}, C1 = {};
        gemm2<256>(r0, r1, bp, C0, C1);

        const float bb = bout[cg];
        #pragma unroll
        for (int rt = 0; rt < 2; ++rt) {
            const v8f& C = rt ? C1 : C0;
            #pragma unroll
            for (int r = 0; r < 8; ++r) {
                const size_t q = (size_t)q0 + (rt << 4) + r + (half << 3);
                out[q * DIM + cg] = C[r] + bb;
            }
        }
    }
}

// ---------------------------------------------------------------------------
extern "C" void kernel_launch(void* const* d_in, const int* in_sizes, int n_in,
                              void* d_out, int out_size, void* d_ws, size_t ws_size,
                              hipStream_t stream)
{
    (void)in_sizes; (void)n_in; (void)out_size; (void)ws_size;

    const float* x     = (const float*)d_in[0];
    const float* vr    = (const float*)d_in[1];
    const float* Wv    = (const float*)d_in[2];
    const float* bv    = (const float*)d_in[3];
    const float* Woff  = (const float*)d_in[4];
    const float* boff  = (const float*)d_in[5];
    const float* Wattn = (const float*)d_in[6];
    const float* battn = (const float*)d_in[7];
    const float* Wout  = (const float*)d_in[8];
    const float* bout  = (const float*)d_in[9];
    // d_in[10] = input_spatial_shapes (compile-time constants here)

    float* ws       = (float*)d_ws;
    float* valueBuf = ws;                                         // 64 MB
    float* locBuf   = valueBuf + (size_t)NQ * DIM;                // 24 MB
    float* attnBuf  = locBuf + (size_t)NQ * HEADS * PTS * 3;      //  8 MB
    float* out      = (float*)d_out;

    proj_kernel<<<NQ / QT, 256, 0, stream>>>(
        x, vr, Wv, bv, Woff, boff, Wattn, battn, valueBuf, locBuf, attnBuf);
    softmax_kernel<<<(NQ * HEADS) / 256, 256, 0, stream>>>(attnBuf);
    sample_out_kernel<<<NQ / QT, 256, 0, stream>>>(
        valueBuf, locBuf, attnBuf, Wout, bout, out);
}